// Global_Feature_Gather_87290915324438
// MI455X (gfx1250) — compile-verified
//
#include <hip/hip_runtime.h>
#include <math.h>

typedef __attribute__((ext_vector_type(2))) float v2f;
typedef __attribute__((ext_vector_type(8))) float v8f;

#define BN   2
#define HH   48
#define WW   48
#define LL   2304           // HH*WW
#define CIN  96
#define DMOD 192
#define DINN 384
#define DST  16
#define DTR  12
#define KK4  4
#define CDBL 44             // DTR + 2*DST

__device__ __forceinline__ float gelu_f(float v) {
    return 0.5f * v * (1.0f + erff(v * 0.70710678118654752f));
}
__device__ __forceinline__ float silu_f(float v) {
    return v / (1.0f + expf(-v));
}

// scan-position t of direction k -> spatial pixel p (h*WW + w)
__device__ __forceinline__ int perm_idx(int k, int t) {
    int tt = (k & 2) ? (LL - 1 - t) : t;
    if (k & 1) return (tt % HH) * WW + (tt / HH);
    return tt;
}

// Accumulate a 16x16 f32 tile with V_WMMA_F32_16X16X4_F32.
// A element (m,k): A[m*ldAm + k*ldAk]; B element (k,n): B[k*ldBk + n*ldBn].
__device__ __forceinline__ v8f wmma_accum_f32(const float* __restrict__ A, int ldAm, int ldAk,
                                              const float* __restrict__ B, int ldBk, int ldBn,
                                              int Kd, v8f c, int row, int hi) {
    for (int kk = 0; kk < Kd; kk += 4) {
        int k0 = kk + 2 * hi;
        v2f a, b;
        a.x = A[row * ldAm + k0 * ldAk];
        a.y = A[row * ldAm + (k0 + 1) * ldAk];
        b.x = B[k0 * ldBk + row * ldBn];
        b.y = B[(k0 + 1) * ldBk + row * ldBn];
        c = __builtin_amdgcn_wmma_f32_16x16x4_f32(false, a, false, b, (short)0, c, false, false);
    }
    return c;
}

// K1: x1[b,p,o] = gelu( sum_c w_init[o,c]*x[b,c,p] + b_init[o] )   (pixel-major out)
__global__ void k_init_proj(const float* __restrict__ x, const float* __restrict__ w,
                            const float* __restrict__ bias, float* __restrict__ x1) {
    int wave = (blockIdx.x * blockDim.x + threadIdx.x) >> 5;
    int lane = threadIdx.x & 31;
    if (wave >= 288 * 12) return;
    int mt = wave / 12, nt = wave % 12;
    int b = mt / 144, p0 = (mt % 144) * 16, n0 = nt * 16;
    int row = lane & 15, hi = lane >> 4;
    const float* A = x + (size_t)b * CIN * LL + p0;   // A[m][k]=x[b][k][p0+m]
    const float* B = w + (size_t)n0 * CIN;            // B[k][n]=w[(n0+n)*96+k]
    v8f c = {};
    c = wmma_accum_f32(A, 1, LL, B, 1, CIN, CIN, c, row, hi);
    int n = n0 + row;
    float bb = bias[n];
    for (int vr = 0; vr < 8; ++vr) {
        int m = p0 + vr + 8 * hi;
        x1[((size_t)b * LL + m) * DMOD + n] = gelu_f(c[vr] + bb);
    }
}

// K2: xz[b,p,o]=sum_c x1[b,p,c]*in_proj_w[o,c]; split into xx (o<384) and z
__global__ void k_inproj(const float* __restrict__ x1, const float* __restrict__ w,
                         float* __restrict__ xx, float* __restrict__ z) {
    int wave = (blockIdx.x * blockDim.x + threadIdx.x) >> 5;
    int lane = threadIdx.x & 31;
    if (wave >= 288 * 48) return;
    int mt = wave / 48, nt = wave % 48;
    int b = mt / 144, p0 = (mt % 144) * 16, n0 = nt * 16;
    int row = lane & 15, hi = lane >> 4;
    const float* A = x1 + ((size_t)b * LL + p0) * DMOD;
    const float* B = w + (size_t)n0 * DMOD;
    v8f c = {};
    c = wmma_accum_f32(A, DMOD, 1, B, 1, DMOD, DMOD, c, row, hi);
    int o = n0 + row;
    float* dst = (o < DINN) ? xx : z;
    int oo = (o < DINN) ? o : (o - DINN);
    for (int vr = 0; vr < 8; ++vr) {
        int m = p0 + vr + 8 * hi;
        dst[((size_t)b * LL + m) * DINN + oo] = c[vr];
    }
}

// K3: depthwise 3x3 SAME conv + bias + silu ; xx(b,p,d) -> xc(b,d,p)
__global__ void k_dwconv(const float* __restrict__ xx, const float* __restrict__ cw,
                         const float* __restrict__ cb, float* __restrict__ xc) {
    int t = blockIdx.x * blockDim.x + threadIdx.x;
    if (t >= BN * DINN * LL) return;
    int d = t % DINN;
    int rest = t / DINN;
    int p = rest % LL, b = rest / LL;
    int h = p / WW, w = p % WW;
    float acc = 0.f;
#pragma unroll
    for (int dh = -1; dh <= 1; ++dh)
#pragma unroll
        for (int dw = -1; dw <= 1; ++dw) {
            int hh = h + dh, ww = w + dw;
            if (hh >= 0 && hh < HH && ww >= 0 && ww < WW)
                acc += cw[d * 9 + (dh + 1) * 3 + (dw + 1)] *
                       xx[((size_t)b * LL + hh * WW + ww) * DINN + d];
        }
    acc += cb[d];
    xc[((size_t)b * DINN + d) * LL + p] = silu_f(acc);
}

// K4: x_dbl[b,k,c,t] = sum_d x_proj_w[k,c,d] * xc[b,d,perm(k,t)]   (WMMA, gathered A)
__global__ void k_xdbl(const float* __restrict__ xc, const float* __restrict__ xpw,
                       float* __restrict__ xdbl) {
    int wave = (blockIdx.x * blockDim.x + threadIdx.x) >> 5;
    int lane = threadIdx.x & 31;
    if (wave >= 8 * 432) return;
    int bk = wave / 432, rem = wave % 432;
    int mt = rem / 3, nt = rem % 3;
    int b = bk >> 2, k = bk & 3;
    int t0 = mt * 16, c0 = nt * 16;
    int row = lane & 15, hi = lane >> 4;
    int p_row = perm_idx(k, t0 + row);
    const float* xcb = xc + (size_t)b * DINN * LL;
    int cc = c0 + row;
    const float* wr = xpw + ((size_t)k * CDBL + (cc < CDBL ? cc : 0)) * DINN;
    float bscale = (cc < CDBL) ? 1.f : 0.f;
    v8f c = {};
    for (int kk = 0; kk < DINN; kk += 4) {
        int k0 = kk + 2 * hi;
        v2f a, bb;
        a.x = xcb[(size_t)k0 * LL + p_row];
        a.y = xcb[(size_t)(k0 + 1) * LL + p_row];
        bb.x = wr[k0] * bscale;
        bb.y = wr[k0 + 1] * bscale;
        c = __builtin_amdgcn_wmma_f32_16x16x4_f32(false, a, false, bb, (short)0, c, false, false);
    }
    if (cc < CDBL) {
        float* orow = xdbl + ((size_t)bk * CDBL + cc) * LL;
        for (int vr = 0; vr < 8; ++vr) orow[t0 + vr + 8 * hi] = c[vr];
    }
}

// K5: selective scan; one thread per (b,k,d); dt GEMM fused; scatter y via atomicAdd
__global__ void k_scan(const float* __restrict__ xc, const float* __restrict__ xdbl,
                       const float* __restrict__ dt_w, const float* __restrict__ dt_b,
                       const float* __restrict__ A_logs, const float* __restrict__ Dsv,
                       float* __restrict__ ysum) {
    int k = blockIdx.y, b = blockIdx.z;
    int d = blockIdx.x * 128 + threadIdx.x;
    int kd = k * DINN + d;
    float An[DST];
#pragma unroll
    for (int n = 0; n < DST; ++n) An[n] = -expf(A_logs[(size_t)kd * DST + n]);
    float dw[DTR];
#pragma unroll
    for (int r = 0; r < DTR; ++r) dw[r] = dt_w[(size_t)kd * DTR + r];
    float dtb = dt_b[kd], Dd = Dsv[kd];
    const float* xcb = xc + ((size_t)b * DINN + d) * LL;
    const float* xdb = xdbl + (size_t)(b * KK4 + k) * CDBL * LL;
    float* yb = ysum + (size_t)b * LL * DINN + d;
    float h[DST];
#pragma unroll
    for (int n = 0; n < DST; ++n) h[n] = 0.f;
    __shared__ float s[CDBL];
    for (int t = 0; t < LL; ++t) {
        __syncthreads();
        if (threadIdx.x < CDBL) {
            s[threadIdx.x] = xdb[(size_t)threadIdx.x * LL + t];
            if (t + 1 < LL) __builtin_prefetch(&xdb[(size_t)threadIdx.x * LL + t + 1], 0, 0);
        }
        __syncthreads();
        int p = perm_idx(k, t);
        float u = xcb[p];
        float dr = dtb;
#pragma unroll
        for (int r = 0; r < DTR; ++r) dr += s[r] * dw[r];
        float delta = (dr > 20.f) ? dr : log1pf(expf(dr));
        float du = delta * u;
        float y = 0.f;
#pragma unroll
        for (int n = 0; n < DST; ++n) {
            h[n] = expf(delta * An[n]) * h[n] + du * s[DTR + n];
            y += h[n] * s[DTR + DST + n];
        }
        y += Dd * u;
        atomicAdd(yb + (size_t)p * DINN, y);
    }
}

// K6: layernorm over D_INNER + silu(z) gate ; one block per pixel
__global__ void k_ln_gate(const float* __restrict__ ysum, const float* __restrict__ zbuf,
                          const float* __restrict__ g, const float* __restrict__ bt,
                          float* __restrict__ out) {
    int pix = blockIdx.x;
    int tid = threadIdx.x;
    const float* yr = ysum + (size_t)pix * DINN;
    float v[3], s1 = 0.f, s2 = 0.f;
#pragma unroll
    for (int i = 0; i < 3; ++i) {
        float vv = yr[tid + i * 128];
        v[i] = vv; s1 += vv; s2 += vv * vv;
    }
    __shared__ float r1[128], r2[128];
    r1[tid] = s1; r2[tid] = s2;
    __syncthreads();
    for (int off = 64; off > 0; off >>= 1) {
        if (tid < off) { r1[tid] += r1[tid + off]; r2[tid] += r2[tid + off]; }
        __syncthreads();
    }
    float mean = r1[0] * (1.f / DINN);
    float var = r2[0] * (1.f / DINN) - mean * mean;
    float inv = rsqrtf(var + 1e-5f);
    const float* zr = zbuf + (size_t)pix * DINN;
    float* orow = out + (size_t)pix * DINN;
#pragma unroll
    for (int i = 0; i < 3; ++i) {
        int dd = tid + i * 128;
        orow[dd] = ((v[i] - mean) * inv * g[dd] + bt[dd]) * silu_f(zr[dd]);
    }
}

// K7: out_proj + residual x1 ; t2[b,p,o] = sum_d yg*W[o,d] + x1[b,p,o]
__global__ void k_outproj(const float* __restrict__ yg, const float* __restrict__ w,
                          const float* __restrict__ x1, float* __restrict__ t2) {
    int wave = (blockIdx.x * blockDim.x + threadIdx.x) >> 5;
    int lane = threadIdx.x & 31;
    if (wave >= 288 * 12) return;
    int mt = wave / 12, nt = wave % 12;
    int b = mt / 144, p0 = (mt % 144) * 16, n0 = nt * 16;
    int row = lane & 15, hi = lane >> 4;
    const float* A = yg + ((size_t)b * LL + p0) * DINN;
    const float* B = w + (size_t)n0 * DINN;
    v8f c = {};
    c = wmma_accum_f32(A, DINN, 1, B, 1, DINN, DINN, c, row, hi);
    int n = n0 + row;
    for (int vr = 0; vr < 8; ++vr) {
        size_t idx = ((size_t)b * LL + p0 + vr + 8 * hi) * DMOD + n;
        t2[idx] = c[vr] + x1[idx];
    }
}

// K8: final conv1x1 + gelu -> d_out (b, 96, h, w)
__global__ void k_final(const float* __restrict__ t2, const float* __restrict__ w,
                        const float* __restrict__ bias, float* __restrict__ out) {
    int wave = (blockIdx.x * blockDim.x + threadIdx.x) >> 5;
    int lane = threadIdx.x & 31;
    if (wave >= 288 * 6) return;
    int mt = wave / 6, nt = wave % 6;
    int b = mt / 144, p0 = (mt % 144) * 16, n0 = nt * 16;
    int row = lane & 15, hi = lane >> 4;
    const float* A = t2 + ((size_t)b * LL + p0) * DMOD;
    const float* B = w + (size_t)n0 * DMOD;
    v8f c = {};
    c = wmma_accum_f32(A, DMOD, 1, B, 1, DMOD, DMOD, c, row, hi);
    int n = n0 + row;
    float bb = bias[n];
    for (int vr = 0; vr < 8; ++vr) {
        int m = p0 + vr + 8 * hi;
        out[((size_t)b * CIN + n) * LL + m] = gelu_f(c[vr] + bb);
    }
}

extern "C" void kernel_launch(void* const* d_in, const int* in_sizes, int n_in,
                              void* d_out, int out_size, void* d_ws, size_t ws_size,
                              hipStream_t stream) {
    const float* x        = (const float*)d_in[0];
    const float* w_init   = (const float*)d_in[1];
    const float* b_init   = (const float*)d_in[2];
    const float* w_fina   = (const float*)d_in[3];
    const float* b_fina   = (const float*)d_in[4];
    const float* in_projw = (const float*)d_in[5];
    const float* conv_w   = (const float*)d_in[6];
    const float* conv_b   = (const float*)d_in[7];
    const float* x_projw  = (const float*)d_in[8];
    const float* dt_w     = (const float*)d_in[9];
    const float* dt_b     = (const float*)d_in[10];
    const float* A_logs   = (const float*)d_in[11];
    const float* Dsv      = (const float*)d_in[12];
    const float* ln_g     = (const float*)d_in[13];
    const float* ln_b     = (const float*)d_in[14];
    const float* out_projw= (const float*)d_in[15];
    float* out = (float*)d_out;

    float* ws = (float*)d_ws;
    const size_t SZ_X1   = (size_t)BN * LL * DMOD;   // 884736
    const size_t SZ_PD   = (size_t)BN * LL * DINN;   // 1769472
    const size_t SZ_XDBL = (size_t)BN * KK4 * CDBL * LL; // 811008
    float* x1   = ws;                       // (b,p,192)
    float* xx   = x1 + SZ_X1;               // (b,p,384)  later reused as gated y
    float* zb   = xx + SZ_PD;               // (b,p,384)  later reused as t2 (b,p,192)
    float* xc   = zb + SZ_PD;               // (b,d,L)
    float* xdbl = xc + SZ_PD;               // (b,k,44,L)
    float* ysum = xdbl + SZ_XDBL;           // (b,p,384)

    k_init_proj<<<432, 256, 0, stream>>>(x, w_init, b_init, x1);
    k_inproj<<<1728, 256, 0, stream>>>(x1, in_projw, xx, zb);
    k_dwconv<<<(BN * DINN * LL + 255) / 256, 256, 0, stream>>>(xx, conv_w, conv_b, xc);
    k_xdbl<<<432, 256, 0, stream>>>(xc, x_projw, xdbl);
    hipMemsetAsync(ysum, 0, SZ_PD * sizeof(float), stream);
    k_scan<<<dim3(DINN / 128, KK4, BN), 128, 0, stream>>>(xc, xdbl, dt_w, dt_b, A_logs, Dsv, ysum);
    k_ln_gate<<<BN * LL, 128, 0, stream>>>(ysum, zb, ln_g, ln_b, xx);   // gated y -> xx
    k_outproj<<<432, 256, 0, stream>>>(xx, out_projw, x1, zb);          // t2 -> zb
    k_final<<<216, 256, 0, stream>>>(zb, w_fina, b_fina, out);
}